// PointSetAttentionWithEdgeBias_70987219468953
// MI455X (gfx1250) — compile-verified
//
#include <hip/hip_runtime.h>
#include <math.h>

typedef __attribute__((ext_vector_type(2))) float v2f;
typedef __attribute__((ext_vector_type(8))) float v8f;

// ---------------- problem constants ----------------
#define NN 10000
#define NE 320000
#define CS 384
#define CV 16
#define CZ 128
#define CZ4 32
#define CH 16
#define H  12
#define PQ 4
#define PV 8
#define A_SCALE  0.8660254037844386f   /* sqrt(H/CH) = sqrt(12/16) */
#define PT_SCALE 0.2041241452319315f   /* sqrt(12/(18*16)) */
#define INV_D    0.1f

// ---------------- workspace layout (floats) ----------------
static const size_t OFF_Q      = 0;                            // N*192
static const size_t OFF_KV     = OFF_Q      + (size_t)NN*192;  // N*384
static const size_t OFF_QPRAW  = OFF_KV     + (size_t)NN*384;  // N*144
static const size_t OFF_KVPRAW = OFF_QPRAW  + (size_t)NN*144;  // N*432
static const size_t OFF_QPTS   = OFF_KVPRAW + (size_t)NN*432;  // N*48*3
static const size_t OFF_KVPTS  = OFF_QPTS   + (size_t)NN*144;  // N*144*3
static const size_t OFF_TQ     = OFF_KVPTS  + (size_t)NN*432;  // N*48*3
static const size_t OFF_TK     = OFF_TQ     + (size_t)NN*144;  // N*48*3
static const size_t OFF_TV     = OFF_TK     + (size_t)NN*144;  // N*96*3
static const size_t OFF_A      = OFF_TV     + (size_t)NN*288;  // E*12
static const size_t OFF_PZ     = OFF_A      + (size_t)NE*12;   // E*32
static const size_t OFF_M      = OFF_PZ     + (size_t)NE*32;   // N*12
static const size_t OFF_DEN    = OFF_M      + (size_t)NN*12;   // N*12
static const size_t OFF_O      = OFF_DEN    + (size_t)NN*12;   // N*192
static const size_t OFF_OPT    = OFF_O      + (size_t)NN*192;  // N*96*3
static const size_t OFF_OPAIR  = OFF_OPT    + (size_t)NN*288;  // N*384
static const size_t OFF_ZERO_END = OFF_OPAIR + (size_t)NN*384; // end of zeroed region
// transposed (K-major) weight copies, written once per launch
static const size_t OFF_BTQ    = OFF_ZERO_END;                 // 192*384
static const size_t OFF_BTKV   = OFF_BTQ    + (size_t)192*384; // 384*384
static const size_t OFF_BTQP   = OFF_BTKV   + (size_t)384*384; // 144*384
static const size_t OFF_BTKVP  = OFF_BTQP   + (size_t)144*384; // 432*384
static const size_t OFF_BTE    = OFF_BTKVP  + (size_t)432*384; // 48*128 (padded [W_b|W_dz|0])
static const size_t OFF_BTOSS  = OFF_BTE    + (size_t)48*128;  // 384*576

// ---------------- weight transpose: BT[c*K + k] = W[k*NC + c] ---------------
__global__ void transpose_w(const float* __restrict__ in, float* __restrict__ out,
                            int K, int NC, int total)
{
    const int i = blockIdx.x * blockDim.x + threadIdx.x;   // k*NC + c
    if (i >= total) return;
    const int k = i / NC, c = i - k * NC;
    out[(size_t)c * K + k] = in[i];
}

// padded edge weight pack: BT_e (48 x 128), cols 0..11 = W_b, 12..43 = W_dz, rest 0
__global__ void pack_edge_bt(const float* __restrict__ Wb, const float* __restrict__ Wdz,
                             float* __restrict__ bt)
{
    const int i = blockIdx.x * blockDim.x + threadIdx.x;   // c*128 + k
    if (i >= 48 * CZ) return;
    const int c = i / CZ, k = i - c * CZ;
    float v = 0.f;
    if (c < H)       v = Wb[k * H + c];
    else if (c < 44) v = Wdz[k * CZ4 + (c - H)];
    bt[i] = v;
}

// ---------------- generic fp32 WMMA GEMM: C = A@B + bias --------------------
// one wave per 16x16 tile; M multiple of 16; NC multiple of 16; BT is K-major.
template <int K, int NC>
__global__ __launch_bounds__(32)
void gemm_wmma_f32(const float* __restrict__ A, const float* __restrict__ BT,
                   const float* __restrict__ bias, float* __restrict__ C)
{
    const int mt = blockIdx.x, nt = blockIdx.y;
    const int lane = threadIdx.x;
    const int r = lane & 15, half = lane >> 4;
    const int row = mt * 16 + r;
    const int col = nt * 16 + r;
    const float* __restrict__ Arow = A  + (size_t)row * K + 2 * half;
    const float* __restrict__ Brow = BT + (size_t)col * K + 2 * half;
    v8f acc = {};
#pragma unroll 8
    for (int k0 = 0; k0 < K; k0 += 4) {
        const v2f a = *(const v2f*)(Arow + k0);
        const v2f b = *(const v2f*)(Brow + k0);
        acc = __builtin_amdgcn_wmma_f32_16x16x4_f32(false, a, false, b,
                                                    (short)0, acc, false, false);
    }
    const float bv = bias[col];
    float* __restrict__ Cp = C + (size_t)(mt * 16 + 8 * half) * NC + col;
#pragma unroll
    for (int g = 0; g < 8; ++g) Cp[(size_t)g * NC] = acc[g] + bv;
}

// ---------------- fused edge GEMM: z @ [W_b | W_dz] (padded 48 cols) --------
__global__ __launch_bounds__(32)
void edge_bias_pairz_wmma(const float* __restrict__ z, const float* __restrict__ bte,
                          const float* __restrict__ bb, const float* __restrict__ bdz,
                          float* __restrict__ a_out, float* __restrict__ pz_out)
{
    const int mt = blockIdx.x;                 // 20000 tiles of 16 edges
    const int lane = threadIdx.x;
    const int r = lane & 15, half = lane >> 4;
    const int row = mt * 16 + r;
    const float* __restrict__ Zrow = z + (size_t)row * CZ + 2 * half;
    const float* __restrict__ B0 = bte + (size_t)(r)      * CZ + 2 * half;
    const float* __restrict__ B1 = bte + (size_t)(16 + r) * CZ + 2 * half;
    const float* __restrict__ B2 = bte + (size_t)(32 + r) * CZ + 2 * half;
    v8f acc0 = {}, acc1 = {}, acc2 = {};
#pragma unroll 8
    for (int k0 = 0; k0 < CZ; k0 += 4) {
        const v2f a = *(const v2f*)(Zrow + k0);
        acc0 = __builtin_amdgcn_wmma_f32_16x16x4_f32(false, a, false, *(const v2f*)(B0 + k0), (short)0, acc0, false, false);
        acc1 = __builtin_amdgcn_wmma_f32_16x16x4_f32(false, a, false, *(const v2f*)(B1 + k0), (short)0, acc1, false, false);
        acc2 = __builtin_amdgcn_wmma_f32_16x16x4_f32(false, a, false, *(const v2f*)(B2 + k0), (short)0, acc2, false, false);
    }
#pragma unroll
    for (int g = 0; g < 8; ++g) {
        const int e = mt * 16 + g + 8 * half;
        // tile0 : logical cols 0..15
        if (r < H) a_out[(size_t)e * H + r] = acc0[g] + bb[r];
        else       pz_out[(size_t)e * CZ4 + (r - H)] = acc0[g] + bdz[r - H];
        // tile1 : logical cols 16..31 -> pair cols 4..19
        pz_out[(size_t)e * CZ4 + (4 + r)] = acc1[g] + bdz[4 + r];
        // tile2 : logical cols 32..43 -> pair cols 20..31
        if (r < 12) pz_out[(size_t)e * CZ4 + (20 + r)] = acc2[g] + bdz[20 + r];
    }
}

// ---------------- rotation: out[n,p,:] = R[n] @ raw[n,:,p]  (raw is (n,3,P)) --
__global__ void rotate_pts(const float* __restrict__ raw, const float* __restrict__ rot,
                           float* __restrict__ out, int P, int total)
{
    const int i = blockIdx.x * blockDim.x + threadIdx.x;   // n*P + p
    if (i >= total) return;
    const int n = i / P, p = i - n * P;
    const float* __restrict__ R = rot + (size_t)n * 9;
    const float* __restrict__ rw = raw + (size_t)n * 3 * P;
    const float x = rw[p], y = rw[P + p], zz = rw[2 * P + p];
    float* __restrict__ o = out + (size_t)i * 3;
    o[0] = R[0] * x + R[1] * y + R[2] * zz;
    o[1] = R[3] * x + R[4] * y + R[5] * zz;
    o[2] = R[6] * x + R[7] * y + R[8] * zz;
}

// ---------------- t_q : (concat[s_v,q_pts] along p) matvec + trans/D ---------
__global__ void tq_kernel(const float* __restrict__ s_v, const float* __restrict__ q_pts,
                          const float* __restrict__ Wtq, const float* __restrict__ trans,
                          float* __restrict__ t_q)
{
    const int i = blockIdx.x * blockDim.x + threadIdx.x;   // n*48 + q
    if (i >= NN * 48) return;
    const int n = i / 48, q = i - n * 48;
    float a0 = trans[n * 3 + 0] * INV_D;
    float a1 = trans[n * 3 + 1] * INV_D;
    float a2 = trans[n * 3 + 2] * INV_D;
#pragma unroll
    for (int p = 0; p < 64; ++p) {
        const float w = Wtq[p * 48 + q];
        const float* in = (p < CV) ? (s_v + ((size_t)n * CV + p) * 3)
                                   : (q_pts + ((size_t)n * 48 + (p - CV)) * 3);
        a0 += in[0] * w; a1 += in[1] * w; a2 += in[2] * w;
    }
    float* o = t_q + (size_t)i * 3;
    o[0] = a0; o[1] = a1; o[2] = a2;
}

// ---------------- t_kv : 160 -> 144 matvec, split into t_k (+trans/D), t_v ---
__global__ void tkv_kernel(const float* __restrict__ s_v, const float* __restrict__ kv_pts,
                           const float* __restrict__ Wtkv, const float* __restrict__ trans,
                           float* __restrict__ t_k, float* __restrict__ t_v)
{
    const int i = blockIdx.x * blockDim.x + threadIdx.x;   // n*144 + q
    if (i >= NN * 144) return;
    const int n = i / 144, q = i - n * 144;
    float a0 = 0.f, a1 = 0.f, a2 = 0.f;
    for (int p = 0; p < 160; ++p) {
        const float w = Wtkv[p * 144 + q];
        const float* in = (p < CV) ? (s_v + ((size_t)n * CV + p) * 3)
                                   : (kv_pts + ((size_t)n * 144 + (p - CV)) * 3);
        a0 += in[0] * w; a1 += in[1] * w; a2 += in[2] * w;
    }
    const int h = q / 12, rr = q - h * 12;
    if (rr < PQ) {
        float* o = t_k + (((size_t)n * 48 + h * PQ + rr)) * 3;
        o[0] = a0 + trans[n * 3 + 0] * INV_D;
        o[1] = a1 + trans[n * 3 + 1] * INV_D;
        o[2] = a2 + trans[n * 3 + 2] * INV_D;
    } else {
        float* o = t_v + (((size_t)n * 96 + h * PV + (rr - PQ))) * 3;
        o[0] = a0; o[1] = a1; o[2] = a2;
    }
}

// ---------------- float atomic max via sign-split bit trick ------------------
__device__ __forceinline__ void atomicMaxFloat(float* addr, float val)
{
    if (val >= 0.f) atomicMax((int*)addr, __float_as_int(val));
    else            atomicMin((unsigned int*)addr, __float_as_uint(val));
}

// ---------------- logits: a += A_SCALE*q.k - hw*||tq-tk||^2 ; seg-max -------
__global__ void logits_kernel(const int* __restrict__ ei,
                              const float* __restrict__ q, const float* __restrict__ kv,
                              const float* __restrict__ t_q, const float* __restrict__ t_k,
                              const float* __restrict__ head_w,
                              float* __restrict__ a, float* __restrict__ m)
{
    const int i = blockIdx.x * blockDim.x + threadIdx.x;   // e*12 + h
    if (i >= NE * H) return;
    const int e = i / H, h = i - e * H;
    const int src = ei[e], dst = ei[NE + e];
    const float* qp = q  + (size_t)dst * (H * CH) + h * CH;
    const float* kp = kv + (size_t)src * (2 * H * CH) + h * (2 * CH);
    float d = 0.f;
#pragma unroll
    for (int c = 0; c < CH; ++c) d += qp[c] * kp[c];
    const float* tqp = t_q + ((size_t)dst * 48 + h * PQ) * 3;
    const float* tkp = t_k + ((size_t)src * 48 + h * PQ) * 3;
    float pt = 0.f;
#pragma unroll
    for (int j = 0; j < PQ * 3; ++j) { const float dl = tqp[j] - tkp[j]; pt += dl * dl; }
    const float hw = log1pf(expf(head_w[h])) * PT_SCALE;
    const float val = a[i] + A_SCALE * d - hw * pt;
    a[i] = val;
    atomicMaxFloat(&m[(size_t)dst * H + h], val);
}

// ---------------- exp + segment denom ---------------------------------------
__global__ void exp_kernel(const int* __restrict__ ei, float* __restrict__ a,
                           const float* __restrict__ m, float* __restrict__ den)
{
    const int i = blockIdx.x * blockDim.x + threadIdx.x;
    if (i >= NE * H) return;
    const int e = i / H, h = i - e * H;
    const int dst = ei[NE + e];
    const float v = expf(a[i] - m[(size_t)dst * H + h]);
    a[i] = v;
    atomicAdd(&den[(size_t)dst * H + h], v);
}

// ---------------- normalized scatter aggregation ----------------------------
__global__ void agg_kernel(const int* __restrict__ ei, const float* __restrict__ a,
                           const float* __restrict__ den, const float* __restrict__ kv,
                           const float* __restrict__ t_v, const float* __restrict__ pz,
                           float* __restrict__ o, float* __restrict__ o_pt,
                           float* __restrict__ o_pair)
{
    const int i = blockIdx.x * blockDim.x + threadIdx.x;
    if (i >= NE * H) return;
    const int e = i / H, h = i - e * H;
    const int src = ei[e], dst = ei[NE + e];
    const float w = a[i] / (den[(size_t)dst * H + h] + 1e-16f);
    const float* vp = kv + (size_t)src * (2 * H * CH) + h * (2 * CH) + CH;
    float* op = o + (size_t)dst * (H * CH) + h * CH;
#pragma unroll
    for (int c = 0; c < CH; ++c) atomicAdd(&op[c], w * vp[c]);
    const float* tvp = t_v + ((size_t)src * 96 + h * PV) * 3;
    float* optp = o_pt + ((size_t)dst * 96 + h * PV) * 3;
#pragma unroll
    for (int j = 0; j < PV * 3; ++j) atomicAdd(&optp[j], w * tvp[j]);
    const float* pzp = pz + (size_t)e * CZ4;
    float* opp = o_pair + (size_t)dst * (H * CZ4) + h * CZ4;
#pragma unroll
    for (int c = 0; c < CZ4; ++c) atomicAdd(&opp[c], w * pzp[c]);
}

// ---------------- final GEMM: [o | o_pair](N,576) @ W_out_ss(576,384) -------
__global__ __launch_bounds__(32)
void final_gemm_wmma(const float* __restrict__ o, const float* __restrict__ o_pair,
                     const float* __restrict__ BT, const float* __restrict__ bias,
                     float* __restrict__ out)
{
    const int mt = blockIdx.x, nt = blockIdx.y;
    const int lane = threadIdx.x;
    const int r = lane & 15, half = lane >> 4;
    const int row = mt * 16 + r;
    const int col = nt * 16 + r;
    const float* __restrict__ Orow = o      + (size_t)row * (H * CH)  + 2 * half;
    const float* __restrict__ Prow = o_pair + (size_t)row * (H * CZ4) + 2 * half;
    const float* __restrict__ Brow = BT + (size_t)col * 576 + 2 * half;
    v8f acc = {};
#pragma unroll 8
    for (int k0 = 0; k0 < 192; k0 += 4) {
        const v2f a = *(const v2f*)(Orow + k0);
        const v2f b = *(const v2f*)(Brow + k0);
        acc = __builtin_amdgcn_wmma_f32_16x16x4_f32(false, a, false, b,
                                                    (short)0, acc, false, false);
    }
#pragma unroll 8
    for (int k0 = 0; k0 < 384; k0 += 4) {
        const v2f a = *(const v2f*)(Prow + k0);
        const v2f b = *(const v2f*)(Brow + 192 + k0);
        acc = __builtin_amdgcn_wmma_f32_16x16x4_f32(false, a, false, b,
                                                    (short)0, acc, false, false);
    }
    const float bv = bias[col];
    float* __restrict__ Cp = out + (size_t)(mt * 16 + 8 * half) * CS + col;
#pragma unroll
    for (int g = 0; g < 8; ++g) Cp[(size_t)g * CS] = acc[g] + bv;
}

// ---------------- out_s_v: o_pt(N,96,3)^T @ W_out_sv(96,16) -----------------
__global__ void outsv_kernel(const float* __restrict__ o_pt,
                             const float* __restrict__ Wsv, float* __restrict__ out)
{
    const int i = blockIdx.x * blockDim.x + threadIdx.x;   // n*48 + cv*3 + d
    if (i >= NN * CV * 3) return;
    const int n = i / 48, rem = i - n * 48;
    const int cv = rem / 3, d = rem - cv * 3;
    float acc = 0.f;
    for (int p = 0; p < H * PV; ++p)
        acc += o_pt[((size_t)n * 96 + p) * 3 + d] * Wsv[p * CV + cv];
    out[i] = acc;
}

// ---------------- launch ----------------------------------------------------
extern "C" void kernel_launch(void* const* d_in, const int* in_sizes, int n_in,
                              void* d_out, int out_size, void* d_ws, size_t ws_size,
                              hipStream_t stream)
{
    const float* s_s    = (const float*)d_in[0];
    const float* s_v    = (const float*)d_in[1];
    const float* rot    = (const float*)d_in[2];
    const float* trans  = (const float*)d_in[3];
    const float* z      = (const float*)d_in[4];
    const int*   ei     = (const int*)  d_in[5];
    const float* W_q    = (const float*)d_in[6];
    const float* b_q    = (const float*)d_in[7];
    const float* W_kv   = (const float*)d_in[8];
    const float* b_kv   = (const float*)d_in[9];
    const float* W_qp   = (const float*)d_in[10];
    const float* b_qp   = (const float*)d_in[11];
    const float* W_kvp  = (const float*)d_in[12];
    const float* b_kvp  = (const float*)d_in[13];
    const float* W_tq   = (const float*)d_in[14];
    const float* W_tkv  = (const float*)d_in[15];
    const float* W_b    = (const float*)d_in[16];
    const float* b_b    = (const float*)d_in[17];
    const float* W_dz   = (const float*)d_in[18];
    const float* b_dz   = (const float*)d_in[19];
    const float* headw  = (const float*)d_in[20];
    const float* W_oss  = (const float*)d_in[21];
    const float* b_oss  = (const float*)d_in[22];
    const float* W_osv  = (const float*)d_in[23];

    float* ws = (float*)d_ws;
    float* q      = ws + OFF_Q;
    float* kv     = ws + OFF_KV;
    float* qpraw  = ws + OFF_QPRAW;
    float* kvpraw = ws + OFF_KVPRAW;
    float* qpts   = ws + OFF_QPTS;
    float* kvpts  = ws + OFF_KVPTS;
    float* t_q    = ws + OFF_TQ;
    float* t_k    = ws + OFF_TK;
    float* t_v    = ws + OFF_TV;
    float* a      = ws + OFF_A;
    float* pz     = ws + OFF_PZ;
    float* m      = ws + OFF_M;
    float* den    = ws + OFF_DEN;
    float* o      = ws + OFF_O;
    float* o_pt   = ws + OFF_OPT;
    float* o_pair = ws + OFF_OPAIR;
    float* btq    = ws + OFF_BTQ;
    float* btkv   = ws + OFF_BTKV;
    float* btqp   = ws + OFF_BTQP;
    float* btkvp  = ws + OFF_BTKVP;
    float* bte    = ws + OFF_BTE;
    float* btoss  = ws + OFF_BTOSS;

    float* out_ss = (float*)d_out;                   // (N, 384)
    float* out_sv = (float*)d_out + (size_t)NN * CS; // (N, 16, 3)

    // init: m <- 0xFFFFFFFF bit pattern (acts as -inf for the atomic trick),
    // den/o/o_pt/o_pair <- 0 (contiguous region)
    hipMemsetAsync(m, 0xFF, (size_t)NN * H * sizeof(float), stream);
    hipMemsetAsync(den, 0, (OFF_ZERO_END - OFF_DEN) * sizeof(float), stream);

    // transpose / pack weights into K-major form (tiny, once per launch)
    transpose_w<<<(384 * 192 + 255) / 256, 256, 0, stream>>>(W_q,   btq,   CS, 192, 384 * 192);
    transpose_w<<<(384 * 384 + 255) / 256, 256, 0, stream>>>(W_kv,  btkv,  CS, 384, 384 * 384);
    transpose_w<<<(384 * 144 + 255) / 256, 256, 0, stream>>>(W_qp,  btqp,  CS, 144, 384 * 144);
    transpose_w<<<(384 * 432 + 255) / 256, 256, 0, stream>>>(W_kvp, btkvp, CS, 432, 384 * 432);
    transpose_w<<<(576 * 384 + 255) / 256, 256, 0, stream>>>(W_oss, btoss, 576, 384, 576 * 384);
    pack_edge_bt<<<(48 * CZ + 255) / 256, 256, 0, stream>>>(W_b, W_dz, bte);

    const int MT = NN / 16;       // 625
    const int ET = NE / 16;       // 20000

    // node projections (fp32 WMMA, branch-free inner loops, b64 frag loads)
    gemm_wmma_f32<CS, 192><<<dim3(MT, 12), 32, 0, stream>>>(s_s, btq,   b_q,   q);
    gemm_wmma_f32<CS, 384><<<dim3(MT, 24), 32, 0, stream>>>(s_s, btkv,  b_kv,  kv);
    gemm_wmma_f32<CS, 144><<<dim3(MT,  9), 32, 0, stream>>>(s_s, btqp,  b_qp,  qpraw);
    gemm_wmma_f32<CS, 432><<<dim3(MT, 27), 32, 0, stream>>>(s_s, btkvp, b_kvp, kvpraw);

    // rotate points into global frame
    rotate_pts<<<(NN * 48 + 255) / 256, 256, 0, stream>>>(qpraw, rot, qpts, 48, NN * 48);
    rotate_pts<<<(NN * 144 + 255) / 256, 256, 0, stream>>>(kvpraw, rot, kvpts, 144, NN * 144);

    // t_q / t_k / t_v
    tq_kernel<<<(NN * 48 + 255) / 256, 256, 0, stream>>>(s_v, qpts, W_tq, trans, t_q);
    tkv_kernel<<<(NN * 144 + 255) / 256, 256, 0, stream>>>(s_v, kvpts, W_tkv, trans, t_k, t_v);

    // edge GEMM: b_edge (into a) + pair_z, single pass over z (fp32 WMMA)
    edge_bias_pairz_wmma<<<ET, 32, 0, stream>>>(z, bte, b_b, b_dz, a, pz);

    // logits + segment max
    logits_kernel<<<(NE * H + 255) / 256, 256, 0, stream>>>(ei, q, kv, t_q, t_k, headw, a, m);

    // exp + segment sum
    exp_kernel<<<(NE * H + 255) / 256, 256, 0, stream>>>(ei, a, m, den);

    // normalized scatter aggregation
    agg_kernel<<<(NE * H + 255) / 256, 256, 0, stream>>>(ei, a, den, kv, t_v, pz,
                                                         o, o_pt, o_pair);

    // outputs
    final_gemm_wmma<<<dim3(MT, 24), 32, 0, stream>>>(o, o_pair, btoss, b_oss, out_ss);
    outsv_kernel<<<(NN * CV * 3 + 255) / 256, 256, 0, stream>>>(o_pt, W_osv, out_sv);
}